// FAM_16441134809583
// MI455X (gfx1250) — compile-verified
//
#include <hip/hip_runtime.h>
#include <hip/hip_bf16.h>

typedef __attribute__((ext_vector_type(2))) float v2f;
typedef __attribute__((ext_vector_type(8))) float v8f;

#define N_BATCH 16
#define C_CH    256
#define HW      128
#define CK      72      // GROUP*K*K = 8*9
#define CG      32      // channels per group
#define SPATIAL 16384   // 128*128

// ---------------------------------------------------------------------------
// Pass 1: pooled[n*C + c] = mean over HW of x[n,c,:,:]
// One block per (n,c) plane: 256 threads x 16 float4 = 16384 floats, coalesced.
// ---------------------------------------------------------------------------
__global__ __launch_bounds__(256) void fam_pool_kernel(
    const float* __restrict__ x, float* __restrict__ pooled)
{
    __shared__ float sred[256];
    const float4* p = (const float4*)(x + (size_t)blockIdx.x * SPATIAL);
    float s = 0.0f;
#pragma unroll
    for (int i = 0; i < 16; ++i) {
        float4 v = p[threadIdx.x + (i << 8)];
        s += v.x + v.y + v.z + v.w;
    }
    sred[threadIdx.x] = s;
    __syncthreads();
    for (int off = 128; off > 0; off >>= 1) {
        if (threadIdx.x < (unsigned)off) sred[threadIdx.x] += sred[threadIdx.x + off];
        __syncthreads();
    }
    if (threadIdx.x == 0) pooled[blockIdx.x] = sred[0] * (1.0f / (float)SPATIAL);
}

// ---------------------------------------------------------------------------
// Pass 2: filt[m, col] = tanh(BN(pooled @ conv_w^T))   (16 x 72)
// WMMA f32 16x16x4: M=16 (batch), N tiles of 16 (cols 0..79, mask >=72),
// K=256 in 64 steps. 1 block x 160 threads = 5 waves, one N-tile per wave.
//
// Per-lane layouts (CDNA5 ISA 7.12.2, 32-bit):
//   A 16x4 : lane l -> row m=l&15, VGPR v holds K = 2*(l>>4) + v
//   B 4x16 : lane l -> col  =l&15, VGPR v holds K = 2*(l>>4) + v
//   C/D    : lane l -> col  =l&15, VGPR v holds row m = v + 8*(l>>4)
// ---------------------------------------------------------------------------
__global__ __launch_bounds__(160) void fam_filt_wmma_kernel(
    const float* __restrict__ pooled, const float* __restrict__ conv_w,
    const float* __restrict__ bn_gamma, const float* __restrict__ bn_beta,
    const float* __restrict__ bn_mean, const float* __restrict__ bn_var,
    float* __restrict__ filt)
{
    const int lane = threadIdx.x & 31;
    const int wave = threadIdx.x >> 5;
    const int m    = lane & 15;             // A row
    const int kh   = (lane >> 4) << 1;      // K sub-offset: 0 or 2
    const int col  = wave * 16 + (lane & 15);
    const int colc = (col < CK) ? col : (CK - 1);
    const float msk = (col < CK) ? 1.0f : 0.0f;

    const float* ap = pooled + m * C_CH + kh;
    const float* bp = conv_w + (size_t)colc * C_CH + kh;

    v8f acc = {};
    for (int kk = 0; kk < C_CH; kk += 4) {
        v2f a, b;
        a.x = ap[kk];     a.y = ap[kk + 1];
        b.x = bp[kk] * msk; b.y = bp[kk + 1] * msk;
        // 8 args: (neg_a, A, neg_b, B, c_mod, C, reuse_a, reuse_b)
        acc = __builtin_amdgcn_wmma_f32_16x16x4_f32(
            false, a, false, b, (short)0, acc, false, false);
    }

    if (col < CK) {
        const float scale = bn_gamma[col] * rsqrtf(bn_var[col] + 1e-5f);
        const float mean  = bn_mean[col];
        const float beta  = bn_beta[col];
        const int mbase = (lane >> 4) << 3;   // 0 or 8
#pragma unroll
        for (int v = 0; v < 8; ++v) {
            float raw = acc[v];
            filt[(mbase + v) * CK + col] = tanhf((raw - mean) * scale + beta);
        }
    }
}

// ---------------------------------------------------------------------------
// Pass 3: fused dynamic 3x3 depthwise conv (reflect pad) + freq recombination.
// Block (32,8): 32 threads x 4px cover one 128-wide row, 8 rows per block.
// Grid (16, 256, 16) = (H/8, C, N). float4 in, float4 out; 2 edge scalars/row.
// ---------------------------------------------------------------------------
__global__ __launch_bounds__(256) void fam_main_kernel(
    const float* __restrict__ x, const float* __restrict__ pooled,
    const float* __restrict__ filt, const float* __restrict__ lamb_l,
    const float* __restrict__ lamb_h, const float* __restrict__ inside_all,
    float* __restrict__ out)
{
    const int n   = blockIdx.z;
    const int ch  = blockIdx.y;
    const int row = blockIdx.x * 8 + threadIdx.y;
    const int c0  = threadIdx.x * 4;

    const float* wt = filt + n * CK + (ch >> 5) * 9;   // group = ch / CG
    const float w0 = wt[0], w1 = wt[1], w2 = wt[2];
    const float w3 = wt[3], w4 = wt[4], w5 = wt[5];
    const float w6 = wt[6], w7 = wt[7], w8 = wt[8];

    const float ia  = inside_all[ch];
    const float ll  = lamb_l[ch] + 1.0f;
    const float lh  = lamb_h[ch] + 1.0f;
    const float gap = pooled[n * C_CH + ch];

    const size_t plane = ((size_t)n * C_CH + ch) << 14;   // *16384
    const float* base  = x + plane;

    // reflect indices (jnp.pad 'reflect': -1 -> 1, H -> H-2)
    const int rm = (row == 0)       ? 1   : row - 1;
    const int rp = (row == HW - 1)  ? 126 : row + 1;
    const int cl = (c0 == 0)        ? 1   : c0 - 1;
    const int cr = (c0 == HW - 4)   ? 126 : c0 + 4;

    const int rr[3] = { rm, row, rp };
    float v[3][6];
#pragma unroll
    for (int d = 0; d < 3; ++d) {
        const float* rb = base + rr[d] * HW;
        float4 f = *(const float4*)(rb + c0);
        v[d][0] = rb[cl];
        v[d][1] = f.x; v[d][2] = f.y; v[d][3] = f.z; v[d][4] = f.w;
        v[d][5] = rb[cr];
    }

    float4 o;
    float* op = (float*)&o;
#pragma unroll
    for (int j = 0; j < 4; ++j) {
        float low = w0 * v[0][j] + w1 * v[0][j + 1] + w2 * v[0][j + 2]
                  + w3 * v[1][j] + w4 * v[1][j + 1] + w5 * v[1][j + 2]
                  + w6 * v[2][j] + w7 * v[2][j + 1] + w8 * v[2][j + 2];
        float xv = v[1][j + 1];
        op[j] = (low * (ia + 1.0f) - ia * gap) * ll + (xv - low) * lh;
    }
    *(float4*)(out + plane + (size_t)row * HW + c0) = o;
}

// ---------------------------------------------------------------------------
extern "C" void kernel_launch(void* const* d_in, const int* in_sizes, int n_in,
                              void* d_out, int out_size, void* d_ws, size_t ws_size,
                              hipStream_t stream) {
    const float* x          = (const float*)d_in[0];
    const float* conv_w     = (const float*)d_in[1];
    const float* bn_gamma   = (const float*)d_in[2];
    const float* bn_beta    = (const float*)d_in[3];
    const float* bn_mean    = (const float*)d_in[4];
    const float* bn_var     = (const float*)d_in[5];
    const float* lamb_l     = (const float*)d_in[6];
    const float* lamb_h     = (const float*)d_in[7];
    const float* inside_all = (const float*)d_in[8];
    float* out = (float*)d_out;

    float* pooled = (float*)d_ws;            // 4096 floats
    float* filt   = pooled + N_BATCH * C_CH; // 1152 floats

    fam_pool_kernel<<<N_BATCH * C_CH, 256, 0, stream>>>(x, pooled);
    fam_filt_wmma_kernel<<<1, 160, 0, stream>>>(pooled, conv_w, bn_gamma, bn_beta,
                                                bn_mean, bn_var, filt);
    fam_main_kernel<<<dim3(HW / 8, C_CH, N_BATCH), dim3(32, 8), 0, stream>>>(
        x, pooled, filt, lamb_l, lamb_h, inside_all, out);
}